// GeometricModule_10703058502028
// MI455X (gfx1250) — compile-verified
//
#include <hip/hip_runtime.h>
#include <math.h>

#define B_  8
#define N_  4096
#define KNN 20

typedef __attribute__((ext_vector_type(2))) float v2f;
typedef __attribute__((ext_vector_type(8))) float v8f;
typedef __attribute__((ext_vector_type(4))) unsigned int u32x4;
typedef __attribute__((ext_vector_type(4))) int i32x4;
typedef __attribute__((ext_vector_type(8))) int i32x8;
typedef int v4i_ __attribute__((vector_size(16)));   // matches builtin param type

// ---------------- CDNA5 feature detection (compile-time) -------------------
#if __has_builtin(__builtin_amdgcn_tensor_load_to_lds)
#define HAVE_TDM 1
#else
#define HAVE_TDM 0
#endif

#if __has_builtin(__builtin_amdgcn_global_load_async_to_lds_b128)
#define HAVE_ASYNC 1
#else
#define HAVE_ASYNC 0
#endif

__device__ __forceinline__ void wait_tensorcnt0() {
#if __has_builtin(__builtin_amdgcn_s_wait_tensorcnt)
  __builtin_amdgcn_s_wait_tensorcnt(0);
#else
  asm volatile("s_wait_tensorcnt 0x0" ::: "memory");
#endif
}

__device__ __forceinline__ void wait_asynccnt0() {
#if __has_builtin(__builtin_amdgcn_s_wait_asynccnt)
  __builtin_amdgcn_s_wait_asynccnt(0);
#else
  asm volatile("s_wait_asynccnt 0x0" ::: "memory");
#endif
}

// D(16x16,f32) = A(16x4,f32) x B(4x16,f32) + C  -- CDNA5 V_WMMA_F32_16X16X4_F32
__device__ __forceinline__ v8f wmma_f32_16x16x4(v2f a, v2f b, v8f c) {
  return __builtin_amdgcn_wmma_f32_16x16x4_f32(false, a, false, b, (short)0, c,
                                               false, false);
}

#if HAVE_TDM
// One-shot TDM DMA: contiguous run of n8 8-byte elements, global -> LDS.
// D# group0: count=1 | lds_addr | global_addr(57b) | type=2 ("image")
// D# group1: data_size=3(8B), tensor_dim0=tile_dim0=n8, dim1=1, stride0=n8
__device__ __forceinline__ void tdm_load_1d_to_lds(const void* gsrc,
                                                   unsigned lds_off,
                                                   unsigned n8) {
  unsigned long long ga = (unsigned long long)gsrc;
  u32x4 g0 = {1u,                                     // count=1 (valid user D#)
              lds_off,                                // LDS byte address
              (unsigned)(ga & 0xffffffffu),           // global_addr[31:0]
              (unsigned)((ga >> 32) & 0x01ffffffu) | 0x80000000u};  // addr[56:32]|type=2
  i32x8 g1 = {(int)0x00030000u,                       // data_size=3 (8 bytes)
              (int)((n8 & 0xffffu) << 16),            // tensor_dim0[15:0]
              (int)(((n8 >> 16) & 0xffffu) | (1u << 16)),  // dim0[31:16] | tensor_dim1=1
              (int)((n8 & 0xffffu) << 16),            // tile_dim0 = n8
              1,                                      // tile_dim1=1, tile_dim2=0
              (int)n8,                                // tensor_dim0_stride[31:0]
              0, 0};
  i32x4 z4 = {0, 0, 0, 0};
#if __has_include(<hip/amd_detail/amd_gfx1250_TDM.h>)
  i32x8 z8 = {0, 0, 0, 0, 0, 0, 0, 0};
  __builtin_amdgcn_tensor_load_to_lds(g0, g1, z4, z4, z8, 0);   // clang-23 lane
#else
  __builtin_amdgcn_tensor_load_to_lds(g0, g1, z4, z4, 0);       // ROCm 7.2 lane
#endif
}
#endif

// ---------------------------------------------------------------------------
// smallest eigenvector of symmetric 3x3 (analytic eigenvalues + cross-product)
// ---------------------------------------------------------------------------
__device__ inline void smallest_evec3(float a00, float a01, float a02,
                                      float a11, float a12, float a22,
                                      float& nx, float& ny, float& nz) {
  float p1 = a01 * a01 + a02 * a02 + a12 * a12;
  if (p1 < 1e-20f) {  // already diagonal: axis of smallest diagonal entry
    nx = 1.f; ny = 0.f; nz = 0.f;
    float m = a00;
    if (a11 < m) { m = a11; nx = 0.f; ny = 1.f; nz = 0.f; }
    if (a22 < m) { nx = 0.f; ny = 0.f; nz = 1.f; }
    return;
  }
  float q  = (a00 + a11 + a22) * (1.0f / 3.0f);
  float b00 = a00 - q, b11 = a11 - q, b22 = a22 - q;
  float p2 = b00 * b00 + b11 * b11 + b22 * b22 + 2.f * p1;
  float p  = sqrtf(p2 * (1.0f / 6.0f));
  float ip = 1.0f / p;
  float c00 = b00 * ip, c01 = a01 * ip, c02 = a02 * ip;
  float c11 = b11 * ip, c12 = a12 * ip, c22 = b22 * ip;
  float detC = c00 * (c11 * c22 - c12 * c12)
             - c01 * (c01 * c22 - c12 * c02)
             + c02 * (c01 * c12 - c11 * c02);
  float r = fminf(1.f, fmaxf(-1.f, 0.5f * detC));
  float phi  = acosf(r) * (1.0f / 3.0f);
  float lmin = q + 2.f * p * cosf(phi + 2.0943951023931953f);  // + 2*pi/3
  float r0x = a00 - lmin, r0y = a01,        r0z = a02;
  float r1x = a01,        r1y = a11 - lmin, r1z = a12;
  float r2x = a02,        r2y = a12,        r2z = a22 - lmin;
  float v0x = r0y * r1z - r0z * r1y, v0y = r0z * r1x - r0x * r1z, v0z = r0x * r1y - r0y * r1x;
  float v1x = r0y * r2z - r0z * r2y, v1y = r0z * r2x - r0x * r2z, v1z = r0x * r2y - r0y * r2x;
  float v2x = r1y * r2z - r1z * r2y, v2y = r1z * r2x - r1x * r2z, v2z = r1x * r2y - r1y * r2x;
  float d0 = v0x * v0x + v0y * v0y + v0z * v0z;
  float d1 = v1x * v1x + v1y * v1y + v1z * v1z;
  float d2 = v2x * v2x + v2y * v2y + v2z * v2z;
  float vx = v0x, vy = v0y, vz = v0z, dm = d0;
  if (d1 > dm) { vx = v1x; vy = v1y; vz = v1z; dm = d1; }
  if (d2 > dm) { vx = v2x; vy = v2y; vz = v2z; dm = d2; }
  if (dm <= 1e-30f) { nx = 0.f; ny = 0.f; nz = 1.f; return; }
  float inv = rsqrtf(dm);
  nx = vx * inv; ny = vy * inv; nz = vz * inv;
}

// ---------------------------------------------------------------------------
// Kernel 1: augmented points (x, y, z, |p|^2)
// ---------------------------------------------------------------------------
__global__ void aug_kernel(const float* __restrict__ pc, float4* __restrict__ aug) {
  int i = blockIdx.x * blockDim.x + threadIdx.x;
  if (i >= B_ * N_) return;
  float x = pc[i * 3 + 0], y = pc[i * 3 + 1], z = pc[i * 3 + 2];
  float4 a; a.x = x; a.y = y; a.z = z; a.w = x * x + y * y + z * z;
  aug[i] = a;
}

// ---------------------------------------------------------------------------
// Kernel 2: WMMA KNN scores + top-20 + geometric features (10 -> padded 12)
// One block (4 waves / 128 threads) per 16 query rows.
// Score(m,n) = |q_n|^2 - 2 p_m . q_n  via A=(-2p,1) x B=(q,|q|^2), K=4 WMMA.
// Batch point stage = one TDM DMA (64 KB, global -> LDS).
// ---------------------------------------------------------------------------
__global__ void __launch_bounds__(128)
knn_feat_kernel(const float4* __restrict__ aug, float* __restrict__ feat) {
  extern __shared__ char smem[];
  float4* s_aug = (float4*)smem;                 // 4096 float4 = 64 KB
  float*  s_sc  = (float*)(smem + N_ * 16);      // 16 x 1024 f32 = 64 KB

  const int  tid  = threadIdx.x;
  const int  lane = tid & 31;
  const int  wave = tid >> 5;
  const bool hi   = (lane >= 16);
  const int  ln   = lane & 15;

  const int b  = blockIdx.x >> 8;                // 256 row-tiles per batch
  const int m0 = (blockIdx.x & 255) * 16;
  const float4* augB = aug + b * N_;

#if HAVE_TDM
  if (tid < 32) {                                // wave 0 drives the DMA
    tdm_load_1d_to_lds(augB, (unsigned)(unsigned long long)s_aug, N_ * 2);
    wait_tensorcnt0();
  }
#else
  for (int i = tid; i < N_; i += 128) {
    __builtin_prefetch(&augB[i + 512], 0, 0);    // global_prefetch_b8
    s_aug[i] = augB[i];
  }
#endif
  __syncthreads();

  // A fragment (constant per block): row m0+ln of (-2x,-2y,-2z,1)
  float4 p4 = s_aug[m0 + ln];
  v2f afrag;
  if (hi) { afrag.x = -2.0f * p4.z; afrag.y = 1.0f; }
  else    { afrag.x = -2.0f * p4.x; afrag.y = -2.0f * p4.y; }

  // running top-K (meaningful for tid < 16 only)
  float bestv[KNN];
  int   besti[KNN];
#pragma unroll
  for (int k = 0; k < KNN; ++k) { bestv[k] = 3.0e38f; besti[k] = 0; }
  float worst = 3.0e38f;
  int   wpos  = 0;

  for (int c = 0; c < 4; ++c) {
    // ---- 16 x 1024 score block via WMMA, 16 column-tiles per wave ----
    for (int t = wave; t < 64; t += 4) {
      int colc = t * 16;
      float4 q = s_aug[c * 1024 + colc + ln];
      v2f bfrag;
      if (hi) { bfrag.x = q.z; bfrag.y = q.w; }
      else    { bfrag.x = q.x; bfrag.y = q.y; }
      v8f acc = {0.f, 0.f, 0.f, 0.f, 0.f, 0.f, 0.f, 0.f};
      acc = wmma_f32_16x16x4(afrag, bfrag, acc);
#pragma unroll
      for (int r = 0; r < 8; ++r) {              // C layout: M=r+8*hi, N=ln
        int M = r + (hi ? 8 : 0);
        s_sc[M * 1024 + colc + ln] = acc[r];
      }
    }
    __syncthreads();

    // ---- per-row streaming top-20 merge ----
    if (tid < 16) {
      const float* rowp = s_sc + tid * 1024;
      int base = c * 1024;
      for (int j = 0; j < 1024; ++j) {
        float v = rowp[j];
        if (v < worst) {
          bestv[wpos] = v; besti[wpos] = base + j;
          worst = bestv[0]; wpos = 0;
#pragma unroll
          for (int k = 1; k < KNN; ++k)
            if (bestv[k] > worst) { worst = bestv[k]; wpos = k; }
        }
      }
    }
    __syncthreads();
  }

  // ---- geometric features for row m0+tid ----
  if (tid < 16) {
    int g = m0 + tid;                            // p4 == s_aug[g] for tid<16
    float sq_p = p4.w;
    float cx = 0.f, cy = 0.f, cz = 0.f, curv = 0.f;
#pragma unroll
    for (int k = 0; k < KNN; ++k) {
      float4 q = s_aug[besti[k]];
      cx += q.x; cy += q.y; cz += q.z;
      curv += sqrtf(fmaxf(bestv[k] + sq_p, 0.f));   // true d^2 = score + |p|^2
    }
    const float inv = 1.0f / (float)KNN;
    cx *= inv; cy *= inv; cz *= inv; curv *= inv;

    float a00 = 0.f, a01 = 0.f, a02 = 0.f, a11 = 0.f, a12 = 0.f, a22 = 0.f;
#pragma unroll
    for (int k = 0; k < KNN; ++k) {
      float4 q = s_aug[besti[k]];
      float dx = q.x - cx, dy = q.y - cy, dz = q.z - cz;
      a00 += dx * dx; a01 += dx * dy; a02 += dx * dz;
      a11 += dy * dy; a12 += dy * dz; a22 += dz * dz;
    }
    float nx, ny, nz;
    smallest_evec3(a00, a01, a02, a11, a12, a22, nx, ny, nz);

    float* f = feat + (size_t)(b * N_ + g) * 12;
    f[0] = p4.x; f[1] = p4.y; f[2] = p4.z;
    f[3] = nx;   f[4] = ny;   f[5] = nz;
    f[6] = curv;
    f[7] = cx;   f[8] = cy;   f[9] = cz;
    f[10] = 0.f; f[11] = 0.f;                    // K-pad to 12 for WMMA
  }
}

// ---------------------------------------------------------------------------
// Kernel 3: fused MLP 10->64->128->256 with f32 WMMA, output (B,256,N)
// 8 waves per block; each wave owns one 16-row M-tile + private LDS h buffers.
// w2 (32 KB) staged into LDS with async global->LDS copies.
// ---------------------------------------------------------------------------
__global__ void __launch_bounds__(256)
mlp_kernel(const float* __restrict__ feat,
           const float* __restrict__ w1, const float* __restrict__ b1,
           const float* __restrict__ w2, const float* __restrict__ b2,
           const float* __restrict__ w3, const float* __restrict__ b3,
           float* __restrict__ out) {
  extern __shared__ char smem[];
  const int  tid  = threadIdx.x;
  const int  lane = tid & 31;
  const int  wave = tid >> 5;
  const bool hi   = (lane >= 16);
  const int  ln   = lane & 15;

  float* h1   = (float*)(smem + (size_t)wave * (16 * 64 + 16 * 128) * 4);
  float* h2   = h1 + 16 * 64;
  float* s_w2 = (float*)(smem + (size_t)8 * (16 * 64 + 16 * 128) * 4);  // 32 KB

  // ---- stage w2 into LDS (async global->LDS on CDNA5) ----
#if HAVE_ASYNC
  for (int i = tid * 4; i < 128 * 64; i += 256 * 4) {
    __builtin_amdgcn_global_load_async_to_lds_b128(
        (__attribute__((address_space(1))) v4i_*)(unsigned long long)(w2 + i),
        (__attribute__((address_space(3))) v4i_*)(unsigned)(unsigned long long)(s_w2 + i),
        0, 0);
  }
  wait_asynccnt0();
#else
  for (int i = tid; i < 128 * 64; i += 256) s_w2[i] = w2[i];
#endif

  const int tile = blockIdx.x * 8 + wave;        // 16-row tile over B*N
  const int r0   = tile * 16;
  const float* fr = feat + (size_t)(r0 + ln) * 12;

  // ---- layer 1: 12(K) -> 64, bias + ReLU ----
  for (int nt = 0; nt < 4; ++nt) {
    v8f acc = {0.f, 0.f, 0.f, 0.f, 0.f, 0.f, 0.f, 0.f};
    int n = nt * 16 + ln;
#pragma unroll
    for (int ks = 0; ks < 3; ++ks) {
      float4 fa = *(const float4*)(fr + ks * 4);
      v2f a; if (hi) { a.x = fa.z; a.y = fa.w; } else { a.x = fa.x; a.y = fa.y; }
      int k0 = ks * 4 + (hi ? 2 : 0);
      v2f bb;
      bb.x = (k0     < 10) ? w1[n * 10 + k0]     : 0.f;
      bb.y = (k0 + 1 < 10) ? w1[n * 10 + k0 + 1] : 0.f;
      acc = wmma_f32_16x16x4(a, bb, acc);
    }
    float bias = b1[n];
#pragma unroll
    for (int r = 0; r < 8; ++r)
      h1[(r + (hi ? 8 : 0)) * 64 + n] = fmaxf(acc[r] + bias, 0.f);
  }
  __syncthreads();                               // also publishes s_w2

  // ---- layer 2: 64 -> 128, bias + ReLU (weights from LDS) ----
  for (int nt = 0; nt < 8; ++nt) {
    v8f acc = {0.f, 0.f, 0.f, 0.f, 0.f, 0.f, 0.f, 0.f};
    int n = nt * 16 + ln;
#pragma unroll 4
    for (int ks = 0; ks < 16; ++ks) {
      int k0 = ks * 4 + (hi ? 2 : 0);
      v2f a;  a.x  = h1[ln * 64 + k0];    a.y  = h1[ln * 64 + k0 + 1];
      v2f bb; bb.x = s_w2[n * 64 + k0];   bb.y = s_w2[n * 64 + k0 + 1];
      acc = wmma_f32_16x16x4(a, bb, acc);
    }
    float bias = b2[n];
#pragma unroll
    for (int r = 0; r < 8; ++r)
      h2[(r + (hi ? 8 : 0)) * 128 + n] = fmaxf(acc[r] + bias, 0.f);
  }
  __syncthreads();

  // ---- layer 3: 128 -> 256, write transposed (B,256,N) ----
  const int bIdx  = r0 >> 12;                    // r0 / 4096
  const int nBase = r0 & (N_ - 1);
  for (int nt = 0; nt < 16; ++nt) {
    v8f acc = {0.f, 0.f, 0.f, 0.f, 0.f, 0.f, 0.f, 0.f};
    int n = nt * 16 + ln;                        // output channel
#pragma unroll 4
    for (int ks = 0; ks < 32; ++ks) {
      int k0 = ks * 4 + (hi ? 2 : 0);
      v2f a;  a.x  = h2[ln * 128 + k0];  a.y  = h2[ln * 128 + k0 + 1];
      v2f bb; bb.x = w3[n * 128 + k0];   bb.y = w3[n * 128 + k0 + 1];
      acc = wmma_f32_16x16x4(a, bb, acc);
    }
    float bias = b3[n];
#pragma unroll
    for (int r = 0; r < 8; ++r) {
      int M = r + (hi ? 8 : 0);
      out[(size_t)bIdx * 256 * N_ + (size_t)n * N_ + (nBase + M)] = acc[r] + bias;
    }
  }
}

// ---------------------------------------------------------------------------
extern "C" void kernel_launch(void* const* d_in, const int* in_sizes, int n_in,
                              void* d_out, int out_size, void* d_ws, size_t ws_size,
                              hipStream_t stream) {
  const float* pc = (const float*)d_in[0];
  // d_in[1] = vis_mask (all-True; unused)
  const float* w1 = (const float*)d_in[2];
  const float* b1 = (const float*)d_in[3];
  const float* w2 = (const float*)d_in[4];
  const float* b2 = (const float*)d_in[5];
  const float* w3 = (const float*)d_in[6];
  const float* b3 = (const float*)d_in[7];
  float* out = (float*)d_out;

  float4* aug  = (float4*)d_ws;                                   // B*N float4
  float*  feat = (float*)((char*)d_ws + (size_t)B_ * N_ * 4 * sizeof(float));

  aug_kernel<<<(B_ * N_) / 256, 256, 0, stream>>>(pc, aug);
  knn_feat_kernel<<<(B_ * N_) / 16, 128,
                    (size_t)(N_ * 16 + 16 * 1024 * 4), stream>>>(aug, feat);
  mlp_kernel<<<(B_ * N_) / (16 * 8), 256,
               (size_t)(8 * (16 * 64 + 16 * 128) * 4 + 128 * 64 * 4), stream>>>(
      feat, w1, b1, w2, b2, w3, b3, out);
}